// PointNetSetAbstraction_7516192768033
// MI455X (gfx1250) — compile-verified
//
#include <hip/hip_runtime.h>
#include <hip/hip_bf16.h>

typedef __attribute__((ext_vector_type(2))) float v2f;
typedef __attribute__((ext_vector_type(8))) float v8f;

#define BB 8
#define NN 8192
#define SS 1024
#define KK 32
#define MM (BB * SS * KK)   // 262144
#define R2C 0.04f
#define EPSC 1e-5f

// ---------------------------------------------------------------------------
// 1) Farthest point sampling: one workgroup per batch, coords in registers,
//    argmax via LDS tree carrying (val, idx, x, y, z) of the best point.
// ---------------------------------------------------------------------------
__global__ __launch_bounds__(1024) void fps_kernel(const float* __restrict__ xyz,
                                                   int* __restrict__ fps_idx) {
    const int b = blockIdx.x;
    const int t = threadIdx.x;
    const size_t xb = (size_t)b * 3 * NN;

    float px[8], py[8], pz[8], dist[8];
#pragma unroll
    for (int i = 0; i < 8; ++i) {
        int p = t + i * 1024;
        px[i] = xyz[xb + p];
        py[i] = xyz[xb + NN + p];
        pz[i] = xyz[xb + 2 * NN + p];
        dist[i] = 1e10f;
    }

    __shared__ float rv[1024], rx[1024], ry[1024], rz[1024];
    __shared__ int   ri[1024];
    __shared__ int   s_far;
    __shared__ float s_cx, s_cy, s_cz;

    if (t == 0) {
        s_far = 0;
        s_cx = xyz[xb];
        s_cy = xyz[xb + NN];
        s_cz = xyz[xb + 2 * NN];
    }
    __syncthreads();

    for (int it = 0; it < SS; ++it) {
        if (t == 0) fps_idx[b * SS + it] = s_far;
        const float cx = s_cx, cy = s_cy, cz = s_cz;

        float bestv = -1.0f, bx = 0.f, by = 0.f, bz = 0.f;
        int besti = 0;
#pragma unroll
        for (int i = 0; i < 8; ++i) {
            float dx = px[i] - cx, dy = py[i] - cy, dz = pz[i] - cz;
            float d = dx * dx + dy * dy + dz * dz;
            dist[i] = fminf(dist[i], d);
            if (dist[i] > bestv) {
                bestv = dist[i]; besti = t + i * 1024;
                bx = px[i]; by = py[i]; bz = pz[i];
            }
        }
        rv[t] = bestv; ri[t] = besti; rx[t] = bx; ry[t] = by; rz[t] = bz;
        __syncthreads();
        for (int off = 512; off > 0; off >>= 1) {
            if (t < off) {
                float v2 = rv[t + off]; int i2 = ri[t + off];
                if (v2 > rv[t] || (v2 == rv[t] && i2 < ri[t])) {
                    rv[t] = v2; ri[t] = i2;
                    rx[t] = rx[t + off]; ry[t] = ry[t + off]; rz[t] = rz[t + off];
                }
            }
            __syncthreads();
        }
        if (t == 0) { s_far = ri[0]; s_cx = rx[0]; s_cy = ry[0]; s_cz = rz[0]; }
        __syncthreads();
    }
}

// ---------------------------------------------------------------------------
// 2) Ball query: one wave32 per sampled point; scan N in order, append the
//    first K in-radius indices (ballot + prefix popcount), pad with first.
// ---------------------------------------------------------------------------
__global__ void ball_query_kernel(const float* __restrict__ xyz,
                                  const int* __restrict__ fps_idx,
                                  int* __restrict__ gi,
                                  float* __restrict__ new_xyz) {
    const int gtid = blockIdx.x * blockDim.x + threadIdx.x;
    const int w = gtid >> 5;
    const int lane = gtid & 31;
    const int b = w >> 10;        // / SS
    const int s = w & (SS - 1);
    const size_t xb = (size_t)b * 3 * NN;

    const int ci = fps_idx[b * SS + s];
    const float cx = xyz[xb + ci];
    const float cy = xyz[xb + NN + ci];
    const float cz = xyz[xb + 2 * NN + ci];
    if (lane == 0) {
        new_xyz[(b * SS + s) * 3 + 0] = cx;
        new_xyz[(b * SS + s) * 3 + 1] = cy;
        new_xyz[(b * SS + s) * 3 + 2] = cz;
    }

    const int base = (b * SS + s) * KK;
    int cnt = 0;
    for (int n0 = 0; n0 < NN && cnt < KK; n0 += 32) {
        const int n = n0 + lane;
        float dx = xyz[xb + n] - cx;
        float dy = xyz[xb + NN + n] - cy;
        float dz = xyz[xb + 2 * NN + n] - cz;
        float d = dx * dx + dy * dy + dz * dz;
        bool inr = (d <= R2C);
        unsigned mask = (unsigned)__ballot(inr);
        int prefix = __popc(mask & ((1u << lane) - 1u));
        int pos = cnt + prefix;
        if (inr && pos < KK) gi[base + pos] = n;
        cnt += __popc(mask);
    }
    int filled = cnt < KK ? cnt : KK;
    // same-wave store->load to same address is kept in order (ISA 7.3)
    int firstIdx = gi[base];
    for (int p = filled + lane; p < KK; p += 32) gi[base + p] = firstIdx;
}

// ---------------------------------------------------------------------------
// 3) Build X0 (channel-major, 68 x M): rows 0..63 = gathered point feats,
//    64..66 = relative xyz, 67 = zero pad (so K-loop can run to 68).
// ---------------------------------------------------------------------------
__global__ void build_x0_kernel(const float* __restrict__ xyz,
                                const float* __restrict__ points,
                                const int* __restrict__ gi,
                                const float* __restrict__ new_xyz,
                                float* __restrict__ X0) {
    const int m = blockIdx.x * blockDim.x + threadIdx.x;
    if (m >= MM) return;
    const int b = m / (SS * KK);
    const int s = (m / KK) & (SS - 1);
    const int idx = gi[m];
    const size_t pb = (size_t)b * 64 * NN;
    for (int c = 0; c < 64; ++c)
        X0[(size_t)c * MM + m] = points[pb + (size_t)c * NN + idx];
    const size_t xb = (size_t)b * 3 * NN;
#pragma unroll
    for (int d = 0; d < 3; ++d)
        X0[(size_t)(64 + d) * MM + m] =
            xyz[xb + (size_t)d * NN + idx] - new_xyz[(b * SS + s) * 3 + d];
    X0[(size_t)67 * MM + m] = 0.0f;
}

// ---------------------------------------------------------------------------
// 4) GEMM: Y(O x M) = W(O x C) * X(Cpad x M) + bias, fp32 WMMA 16x16x4.
//    W staged in LDS once per block, zero-padded to CPAD -> the k-loop is
//    pure ds_load + global_load + v_wmma with no exec-mask branching.
//    A-frag: all lanes hold row = lane&15; K = 2*(lane>=16)+{0,1}.
//    B-frag: N = lane&15; K = 2*(lane>=16)+{0,1}.
// ---------------------------------------------------------------------------
template <int OTILES, int CPAD>
__global__ __launch_bounds__(256) void gemm_kernel(const float* __restrict__ W,
                                                   const float* __restrict__ bias,
                                                   const float* __restrict__ X,
                                                   float* __restrict__ Y,
                                                   int Cact) {
    __shared__ float Wlds[OTILES * 16 * CPAD];
    const int tid = threadIdx.x;
    for (int i = tid; i < OTILES * 16 * CPAD; i += 256) {
        const int row = i / CPAD;
        const int c = i - row * CPAD;
        Wlds[i] = (c < Cact) ? W[row * Cact + c] : 0.0f;
    }
    __syncthreads();

    const int gtid = blockIdx.x * blockDim.x + tid;
    const int wave = gtid >> 5;
    const int lane = gtid & 31;
    const int half = lane >> 4;
    const int col = lane & 15;
    const int m0 = wave * 16;

    v8f acc[OTILES] = {};

    for (int kk = 0; kk < CPAD; kk += 4) {
        const int k0 = kk + half * 2;
        v2f bfrag;
        bfrag.x = X[(size_t)(k0 + 0) * MM + m0 + col];
        bfrag.y = X[(size_t)(k0 + 1) * MM + m0 + col];
#pragma unroll
        for (int t = 0; t < OTILES; ++t) {
            const int row = t * 16 + col;
            v2f afrag;
            afrag.x = Wlds[row * CPAD + k0 + 0];
            afrag.y = Wlds[row * CPAD + k0 + 1];
            acc[t] = __builtin_amdgcn_wmma_f32_16x16x4_f32(
                false, afrag, false, bfrag, (short)0, acc[t], false, false);
        }
    }
#pragma unroll
    for (int t = 0; t < OTILES; ++t) {
#pragma unroll
        for (int r = 0; r < 8; ++r) {
            const int o = t * 16 + half * 8 + r;
            Y[(size_t)o * MM + m0 + col] = acc[t][r] + bias[o];
        }
    }
}

// ---------------------------------------------------------------------------
// 5) BatchNorm stats: one block per channel -> mean, rsqrt(var+eps)
// ---------------------------------------------------------------------------
__global__ void bn_stats_kernel(const float* __restrict__ Y,
                                float* __restrict__ stats) {
    const int o = blockIdx.x;
    const int t = threadIdx.x;
    const float* row = Y + (size_t)o * MM;
    float s = 0.f, s2 = 0.f;
    for (int m = t; m < MM; m += 256) {
        float v = row[m];
        s += v; s2 += v * v;
    }
    __shared__ float a[256], a2[256];
    a[t] = s; a2[t] = s2;
    __syncthreads();
    for (int off = 128; off > 0; off >>= 1) {
        if (t < off) { a[t] += a[t + off]; a2[t] += a2[t + off]; }
        __syncthreads();
    }
    if (t == 0) {
        float mean = a[0] / (float)MM;
        float var = a2[0] / (float)MM - mean * mean;
        stats[o * 2 + 0] = mean;
        stats[o * 2 + 1] = rsqrtf(var + EPSC);
    }
}

// ---------------------------------------------------------------------------
// 6) BN apply + ReLU (in place)
// ---------------------------------------------------------------------------
__global__ void bn_apply_kernel(float* __restrict__ Y,
                                const float* __restrict__ stats,
                                const float* __restrict__ g,
                                const float* __restrict__ be,
                                int OM) {
    const int i = blockIdx.x * blockDim.x + threadIdx.x;
    if (i >= OM) return;
    const int o = i / MM;
    const float mean = stats[o * 2 + 0];
    const float rstd = stats[o * 2 + 1];
    float v = Y[i];
    v = g[o] * (v - mean) * rstd + be[o];
    Y[i] = v > 0.f ? v : 0.f;
}

// ---------------------------------------------------------------------------
// 7) Outputs: new_xyz transposed to (B,3,S); max over K -> (B,128,S)
// ---------------------------------------------------------------------------
__global__ void write_newxyz_kernel(const float* __restrict__ new_xyz,
                                    float* __restrict__ out) {
    const int i = blockIdx.x * blockDim.x + threadIdx.x;
    if (i >= BB * 3 * SS) return;
    const int b = i / (3 * SS);
    const int d = (i / SS) % 3;
    const int s = i % SS;
    out[i] = new_xyz[(b * SS + s) * 3 + d];
}

__global__ void max_over_k_kernel(const float* __restrict__ X,
                                  float* __restrict__ out) {
    const int i = blockIdx.x * blockDim.x + threadIdx.x;
    if (i >= BB * 128 * SS) return;
    const int b = i / (128 * SS);
    const int o = (i / SS) % 128;
    const int s = i % SS;
    const size_t base = (size_t)o * MM + (size_t)(b * SS + s) * KK;
    float mx = X[base];
#pragma unroll
    for (int k = 1; k < KK; ++k) mx = fmaxf(mx, X[base + k]);
    out[i] = mx;
}

// ---------------------------------------------------------------------------
// Launch
// ---------------------------------------------------------------------------
extern "C" void kernel_launch(void* const* d_in, const int* in_sizes, int n_in,
                              void* d_out, int out_size, void* d_ws, size_t ws_size,
                              hipStream_t stream) {
    const float* xyz    = (const float*)d_in[0];
    const float* points = (const float*)d_in[1];
    const float* w0 = (const float*)d_in[2];
    const float* b0 = (const float*)d_in[3];
    const float* g0 = (const float*)d_in[4];
    const float* be0 = (const float*)d_in[5];
    const float* w1 = (const float*)d_in[6];
    const float* b1 = (const float*)d_in[7];
    const float* g1 = (const float*)d_in[8];
    const float* be1 = (const float*)d_in[9];
    const float* w2 = (const float*)d_in[10];
    const float* b2 = (const float*)d_in[11];
    const float* g2 = (const float*)d_in[12];
    const float* be2 = (const float*)d_in[13];

    char* ws = (char*)d_ws;
    int*   fps_idx = (int*)(ws + 0);                           // 32 KB
    float* new_xyz = (float*)(ws + 32768);                     // 96 KB
    int*   gi      = (int*)(ws + 131072);                      // 1 MB
    float* stats   = (float*)(ws + 1179648);                   // 1 KB
    float* bufA    = (float*)(ws + 1180672);                   // 68  x M floats
    float* bufB    = (float*)(ws + 1180672 + (size_t)68 * MM * 4); // 128 x M floats

    float* out_xyz  = (float*)d_out;                // B*3*S
    float* out_feat = out_xyz + BB * 3 * SS;        // B*128*S

    fps_kernel<<<BB, 1024, 0, stream>>>(xyz, fps_idx);
    ball_query_kernel<<<(BB * SS) / 8, 256, 0, stream>>>(xyz, fps_idx, gi, new_xyz);
    build_x0_kernel<<<MM / 256, 256, 0, stream>>>(xyz, points, gi, new_xyz, bufA);

    const int gemm_blocks = (MM / 16) * 32 / 256;   // 2048

    // Layer 0: 68-pad x M -> 64 x M
    gemm_kernel<4, 68><<<gemm_blocks, 256, 0, stream>>>(w0, b0, bufA, bufB, 67);
    bn_stats_kernel<<<64, 256, 0, stream>>>(bufB, stats);
    bn_apply_kernel<<<(64 * MM) / 256, 256, 0, stream>>>(bufB, stats, g0, be0, 64 * MM);

    // Layer 1: 64 x M -> 64 x M
    gemm_kernel<4, 64><<<gemm_blocks, 256, 0, stream>>>(w1, b1, bufB, bufA, 64);
    bn_stats_kernel<<<64, 256, 0, stream>>>(bufA, stats);
    bn_apply_kernel<<<(64 * MM) / 256, 256, 0, stream>>>(bufA, stats, g1, be1, 64 * MM);

    // Layer 2: 64 x M -> 128 x M
    gemm_kernel<8, 64><<<gemm_blocks, 256, 0, stream>>>(w2, b2, bufA, bufB, 64);
    bn_stats_kernel<<<128, 256, 0, stream>>>(bufB, stats);
    bn_apply_kernel<<<(128 * MM) / 256, 256, 0, stream>>>(bufB, stats, g2, be2, 128 * MM);

    write_newxyz_kernel<<<(BB * 3 * SS) / 256, 256, 0, stream>>>(new_xyz, out_xyz);
    max_over_k_kernel<<<(BB * 128 * SS) / 256, 256, 0, stream>>>(bufB, out_feat);
}